// FermiLayer_84756884619315
// MI455X (gfx1250) — compile-verified
//
#include <hip/hip_runtime.h>
#include <math.h>

typedef __attribute__((ext_vector_type(2))) float v2f;
typedef __attribute__((ext_vector_type(8))) float v8f;

#define GAIN 1.5927812698663017f
#define INV_SQRT2 0.70710678118654752f

// ---------- small reductions for the h_one path ----------

// gvec[spin*256 + j] = mean over 256 electrons of h_one[:, j]
__global__ void k_reduce_hone(const float* __restrict__ h_one, float* __restrict__ gvec) {
    int spin = blockIdx.x;          // 0 = up, 1 = down
    int j = threadIdx.x;            // 0..255
    const float* base = h_one + (size_t)spin * 256 * 256;
    float s = 0.f;
    for (int i = 0; i < 256; ++i) s += base[(size_t)i * 256 + j];
    gvec[spin * 256 + j] = s * (1.0f / 256.0f);
}

// bias_eff[j] = b1[j] + sum_i gvec[i] * Wg[i][j]   (rank-1 global term folded into bias)
__global__ void k_bias(const float* __restrict__ b1, const float* __restrict__ Wg,
                       const float* __restrict__ gvec, float* __restrict__ bias_eff) {
    int j = threadIdx.x;            // 0..255
    float s = b1[j];
    for (int i = 0; i < 512; ++i) s += gvec[i] * Wg[(size_t)i * 256 + j];
    bias_eff[j] = s;
}

// g2[spin][r][p] = mean over 256 senders of h_two[sender, r, p]
__global__ void k_reduce_htwo(const float* __restrict__ h_two, float* __restrict__ g2) {
    int r = blockIdx.x;             // 0..511 receiver
    int spin = blockIdx.y;          // 0..1
    int p = threadIdx.x;            // 0..127
    const float* base = h_two + (size_t)spin * 256 * 65536 + (size_t)r * 128 + p;
    float s = 0.f;
    for (int i = 0; i < 256; ++i) s += base[(size_t)i * 65536];
    g2[((size_t)spin * 512 + r) * 128 + p] = s * (1.0f / 256.0f);
}

// one_in = concat([h_one, g2_up, g2_dn], axis=1)  -> (512, 512)
__global__ void k_build_onein(const float* __restrict__ h_one, const float* __restrict__ g2,
                              float* __restrict__ one_in) {
    int idx = blockIdx.x * 256 + threadIdx.x;    // 0..262143
    int r = idx >> 9;
    int c = idx & 511;
    float v;
    if (c < 256)       v = h_one[(size_t)r * 256 + c];
    else if (c < 384)  v = g2[(size_t)r * 128 + (c - 256)];
    else               v = g2[(size_t)65536 + (size_t)r * 128 + (c - 384)];
    one_in[idx] = v;
}

// ---------- WMMA GEMM: h_one path  (M=512, K=512, N=256) ----------
__global__ void __launch_bounds__(256) k_gemm_one(
        const float* __restrict__ one_in, const float* __restrict__ W1,
        const float* __restrict__ bias_eff, const float* __restrict__ h_one,
        float* __restrict__ out1) {
    int tid = threadIdx.x;
    int lane = tid & 31;
    int laneLow = lane & 15;
    int laneHi  = lane >> 4;
    int rtile = blockIdx.x * 8 + (tid >> 5);   // 0..31 (16-row tiles)
    int ntile = blockIdx.y;                    // 0..15 (16-col tiles)
    int m0 = rtile * 16;
    int col = ntile * 16 + laneLow;

    const float* arow = one_in + (size_t)(m0 + laneLow) * 512;
    v8f acc = {};
    for (int k = 0; k < 128; ++k) {
        int kb = 4 * k + 2 * laneHi;                   // per-lane K base
        v2f a = *(const v2f*)(arow + kb);              // A: (M=laneLow, K=kb,kb+1)
        v2f b;
        b.x = W1[(size_t)kb * 256 + col];              // B: (K=kb, N=col)
        b.y = W1[(size_t)(kb + 1) * 256 + col];
        acc = __builtin_amdgcn_wmma_f32_16x16x4_f32(false, a, false, b,
                                                    (short)0, acc, false, false);
    }
    float bias = bias_eff[col];
    #pragma unroll
    for (int v = 0; v < 8; ++v) {
        int row = m0 + v + 8 * laneHi;                 // C/D layout
        float x = (acc[v] + bias) * INV_SQRT2;
        float val = tanhf(x) * GAIN;
        size_t idx = (size_t)row * 256 + col;
        out1[idx] = (h_one[idx] + val) * INV_SQRT2;
    }
}

// ---------- WMMA GEMM: h_two path (M=262144, K=128, N=128) + fused act/residual ----------
// A-tiles staged into LDS with async global->LDS copies; B fragments pre-swizzled in LDS.
#define AROW_F 132      // row stride in floats (128 + 4 pad -> 528 B, bank-conflict-free)
__global__ void __launch_bounds__(256) k_gemm_two(
        const float* __restrict__ h_two, const float* __restrict__ W2,
        const float* __restrict__ b2, float* __restrict__ out2) {
    __shared__ v2f   bfrag[8192];          // 64 KB: [ntile][k][lane] = (W2[kb][col], W2[kb+1][col])
    __shared__ float atile[8][16 * AROW_F]; // 66 KB: per-wave 16x128 tile, padded rows

    int tid  = threadIdx.x;
    int lane = tid & 31;
    int wave = tid >> 5;
    size_t rowtile = (size_t)blockIdx.x * 8 + wave;    // 16384 tiles total
    size_t m0 = rowtile * 16;

    // --- async-copy this wave's A tile: 16 rows x 512 B, one row per instruction ---
    // LDS destination address comes from a VGPR (workgroup-relative byte offset = low 32 bits of generic ptr)
    #pragma unroll
    for (int r = 0; r < 16; ++r) {
        unsigned ldsoff = (unsigned)(uintptr_t)&atile[wave][r * AROW_F + lane * 4];
        unsigned long long gaddr =
            (unsigned long long)(uintptr_t)(h_two + (m0 + r) * 128 + lane * 4);
        asm volatile("global_load_async_to_lds_b128 %0, %1, off"
                     :: "v"(ldsoff), "v"(gaddr) : "memory");
    }

    // --- cooperatively build B fragments while the async engine streams A ---
    for (int idx = tid; idx < 8192; idx += 256) {
        int ln = idx & 31;
        int k  = (idx >> 5) & 31;
        int nt = idx >> 10;
        int col = nt * 16 + (ln & 15);
        int kb  = 4 * k + 2 * (ln >> 4);
        v2f b;
        b.x = W2[kb * 128 + col];
        b.y = W2[(kb + 1) * 128 + col];
        bfrag[idx] = b;
    }
    __syncthreads();
    asm volatile("s_wait_asynccnt 0" ::: "memory");    // this wave's A tile is in LDS

    int laneLow = lane & 15;
    int laneHi  = lane >> 4;

    v8f acc[8] = {};
    for (int k = 0; k < 32; ++k) {
        int kb = 4 * k + 2 * laneHi;
        v2f a = *(const v2f*)&atile[wave][laneLow * AROW_F + kb];   // ds_load_b64, no conflicts
        #pragma unroll
        for (int nt = 0; nt < 8; ++nt) {
            v2f b = bfrag[(nt << 10) + (k << 5) + lane];
            acc[nt] = __builtin_amdgcn_wmma_f32_16x16x4_f32(false, a, false, b,
                                                            (short)0, acc[nt], false, false);
        }
    }

    #pragma unroll
    for (int nt = 0; nt < 8; ++nt) {
        int col = nt * 16 + laneLow;
        float bias = b2[col];
        #pragma unroll
        for (int v = 0; v < 8; ++v) {
            int rl = v + 8 * laneHi;                   // C/D layout row within tile
            float x = acc[nt][v] + bias;
            float val = tanhf(x) * GAIN;
            float orig = atile[wave][rl * AROW_F + col];   // residual from LDS (no L2 re-read)
            out2[(m0 + rl) * 128 + col] = (orig + val) * INV_SQRT2;
        }
    }
}

// ---------- launcher ----------
extern "C" void kernel_launch(void* const* d_in, const int* in_sizes, int n_in,
                              void* d_out, int out_size, void* d_ws, size_t ws_size,
                              hipStream_t stream) {
    const float* h_one = (const float*)d_in[0];   // (512, 256)
    const float* h_two = (const float*)d_in[1];   // (512, 512, 128)
    const float* W1    = (const float*)d_in[2];   // (512, 256)
    const float* b1    = (const float*)d_in[3];   // (256,)
    const float* Wg    = (const float*)d_in[4];   // (512, 256)
    const float* W2    = (const float*)d_in[5];   // (128, 128)
    const float* b2    = (const float*)d_in[6];   // (128,)

    float* out1 = (float*)d_out;                  // (512, 256)
    float* out2 = out1 + 512 * 256;               // (512, 512, 128)

    float* ws       = (float*)d_ws;
    float* gvec     = ws;                         // 512
    float* bias_eff = ws + 512;                   // 256
    float* g2       = ws + 768;                   // 2*512*128 = 131072
    float* one_in   = ws + 768 + 131072;          // 512*512  = 262144

    k_reduce_hone<<<2, 256, 0, stream>>>(h_one, gvec);
    k_bias<<<1, 256, 0, stream>>>(b1, Wg, gvec, bias_eff);
    k_reduce_htwo<<<dim3(512, 2), 128, 0, stream>>>(h_two, g2);
    k_build_onein<<<1024, 256, 0, stream>>>(h_one, g2, one_in);
    k_gemm_one<<<dim3(4, 16), 256, 0, stream>>>(one_in, W1, bias_eff, h_one, out1);
    k_gemm_two<<<2048, 256, 0, stream>>>(h_two, W2, b2, out2);
}